// GATConv_5059471475171
// MI455X (gfx1250) — compile-verified
//
#include <hip/hip_runtime.h>
#include <math.h>

typedef __attribute__((ext_vector_type(2))) float v2f;
typedef __attribute__((ext_vector_type(8))) float v8f;

#define IN_C   128
#define OUT_C  128
#define NEG_SLOPE 0.2f
#define ENC_NEG_INF 0x007FFFFFu   // fenc(-inf)

// ---------- monotone float <-> uint encoding for atomicMax on floats ----------
__device__ __forceinline__ unsigned fenc(float f) {
    int b = __float_as_int(f);
    return (b < 0) ? ~(unsigned)b : ((unsigned)b | 0x80000000u);
}
__device__ __forceinline__ float fdec(unsigned u) {
    int b = (u & 0x80000000u) ? (int)(u & 0x7FFFFFFFu) : ~(int)u;
    return __int_as_float(b);
}
__device__ __forceinline__ float readfirst_f(float v) {
    return __int_as_float(__builtin_amdgcn_readfirstlane(__float_as_int(v)));
}

// ---------- P0: zero d_out, init per-node seg-max / seg-sum ----------
__global__ void k_init(float* __restrict__ out, float* __restrict__ sden,
                       unsigned* __restrict__ menc, int Nn) {
    long long idx = (long long)blockIdx.x * blockDim.x + threadIdx.x;
    long long tot = (long long)Nn * OUT_C;
    if (idx < tot) out[idx] = 0.0f;
    if (idx < Nn) { sden[idx] = 0.0f; menc[idx] = ENC_NEG_INF; }
}

// ---------- P1: h = x @ W via V_WMMA_F32_16X16X4_F32 ----------
// One wave computes a 16x64 strip (4 N-tiles, 4 accumulators) so the A b64
// load is reused by 4 WMMAs per k-step. 2 waves cover N=128 for one M block.
// A (16x4 f32): VGPR0 = K=0 (lanes 0-15) / K=2 (lanes 16-31); VGPR1 = K=1 / K=3.
// B (4x16 f32): VGPR0 = row K=0 / K=2; VGPR1 = K=1 / K=3.
// C/D (16x16 f32): VGPR v -> M=v (lanes 0-15), M=v+8 (lanes 16-31), N = lane&15.
__global__ void k_gemm_wmma(const float* __restrict__ x, const float* __restrict__ W,
                            float* __restrict__ h, int Nn) {
    const int wave = threadIdx.x >> 5;
    const int lane = threadIdx.x & 31;
    long long gw = (long long)blockIdx.x * 8 + wave;    // global wave id
    int mt = (int)(gw >> 1);                            // M tile (16 rows)
    int nh = (int)(gw & 1);                             // N half (0: 0-63, 1: 64-127)
    int m0 = mt * 16;
    if (m0 >= Nn) return;

    const int rsub  = lane & 15;     // M row within tile / N col within tile
    const int khalf = lane >> 4;     // 0: K pair {0,1}; 1: K pair {2,3}

    int arow = m0 + rsub; if (arow >= Nn) arow = Nn - 1;
    const float* __restrict__ xrow = x + (long long)arow * IN_C;
    const int ncol = nh * 64 + rsub; // base column this lane touches in W/h

    v8f acc0 = {}, acc1 = {}, acc2 = {}, acc3 = {};
    #pragma unroll 4
    for (int k = 0; k < IN_C; k += 4) {
        v2f a;
        a.x = xrow[k + 2 * khalf];
        a.y = xrow[k + 2 * khalf + 1];
        const float* __restrict__ w0 = W + (long long)(k + 2 * khalf) * OUT_C + ncol;
        const float* __restrict__ w1 = w0 + OUT_C;
        v2f b0 = { w0[0],  w1[0]  };
        v2f b1 = { w0[16], w1[16] };
        v2f b2 = { w0[32], w1[32] };
        v2f b3 = { w0[48], w1[48] };
        acc0 = __builtin_amdgcn_wmma_f32_16x16x4_f32(false, a, false, b0, (short)0, acc0, false, false);
        acc1 = __builtin_amdgcn_wmma_f32_16x16x4_f32(false, a, false, b1, (short)0, acc1, false, false);
        acc2 = __builtin_amdgcn_wmma_f32_16x16x4_f32(false, a, false, b2, (short)0, acc2, false, false);
        acc3 = __builtin_amdgcn_wmma_f32_16x16x4_f32(false, a, false, b3, (short)0, acc3, false, false);
    }

    const int mbase = m0 + khalf * 8;
    if (m0 + 15 < Nn) {               // fast path: full tile, no per-row guards
        #pragma unroll
        for (int v = 0; v < 8; ++v) {
            float* __restrict__ hrow = h + (long long)(mbase + v) * OUT_C + ncol;
            hrow[0]  = acc0[v];
            hrow[16] = acc1[v];
            hrow[32] = acc2[v];
            hrow[48] = acc3[v];
        }
    } else {                          // tail tile
        #pragma unroll
        for (int v = 0; v < 8; ++v) {
            int m = mbase + v;
            if (m < Nn) {
                float* __restrict__ hrow = h + (long long)m * OUT_C + ncol;
                hrow[0]  = acc0[v];
                hrow[16] = acc1[v];
                hrow[32] = acc2[v];
                hrow[48] = acc3[v];
            }
        }
    }
}

// ---------- P2: per-node attention scalars al = h.att_l, ar = h.att_r ----------
__global__ void k_scores(const float* __restrict__ h, const float* __restrict__ att,
                         float* __restrict__ al, float* __restrict__ ar, int Nn) {
    int wid  = blockIdx.x * (blockDim.x >> 5) + (threadIdx.x >> 5);
    int lane = threadIdx.x & 31;
    if (wid >= Nn) return;
    float4 hv = ((const float4*)(h + (long long)wid * OUT_C))[lane];
    float4 a1 = ((const float4*)att)[lane];                 // att[:, :C]
    float4 a2 = ((const float4*)(att + OUT_C))[lane];       // att[:, C:]
    float s1 = hv.x * a1.x + hv.y * a1.y + hv.z * a1.z + hv.w * a1.w;
    float s2 = hv.x * a2.x + hv.y * a2.y + hv.z * a2.z + hv.w * a2.w;
    #pragma unroll
    for (int off = 16; off > 0; off >>= 1) {
        s1 += __shfl_xor(s1, off, 32);
        s2 += __shfl_xor(s2, off, 32);
    }
    if (lane == 0) { al[wid] = s1; ar[wid] = s2; }
}

// ---------- P3: edge logits + segment max ----------
__global__ void k_edge_alpha(const float* __restrict__ al, const float* __restrict__ ar,
                             const int* __restrict__ src, const int* __restrict__ dst,
                             float* __restrict__ aE, unsigned* __restrict__ menc, int Etot) {
    int e = blockIdx.x * blockDim.x + threadIdx.x;
    if (e >= Etot) return;
    int i = dst[e], j = src[e];
    float v = al[i] + ar[j];
    v = (v > 0.0f) ? v : NEG_SLOPE * v;      // leaky_relu
    aE[e] = v;
    atomicMax(&menc[i], fenc(v));
}

// ---------- P4: exp(alpha - max) + segment sum ----------
__global__ void k_edge_exp(const int* __restrict__ dst, float* __restrict__ aE,
                           const unsigned* __restrict__ menc, float* __restrict__ sden,
                           int Etot) {
    int e = blockIdx.x * blockDim.x + threadIdx.x;
    if (e >= Etot) return;
    int i = dst[e];
    float a = expf(aE[e] - fdec(menc[i]));
    aE[e] = a;
    unsafeAtomicAdd(&sden[i], a);
}

// ---------- P5: weighted scatter-add (heavy pass; one wave per edge) ----------
// src/dst/aE are wave-uniform -> readfirstlane so h/out addressing uses SGPR
// bases (saddr-form vmem) with just a lane*16 vector offset.
__global__ void k_edge_scatter(const float* __restrict__ h, const float* __restrict__ aE,
                               const int* __restrict__ src, const int* __restrict__ dst,
                               float* __restrict__ out, int Etot) {
    int wid  = blockIdx.x * (blockDim.x >> 5) + (threadIdx.x >> 5);
    int lane = threadIdx.x & 31;
    if (wid >= Etot) return;
    int   j = __builtin_amdgcn_readfirstlane(src[wid]);
    int   i = __builtin_amdgcn_readfirstlane(dst[wid]);
    float a = readfirst_f(aE[wid]);
    float4 v = ((const float4*)(h + (long long)j * OUT_C))[lane];   // L2-resident gather
    float* o = out + (long long)i * OUT_C + lane * 4;
    unsafeAtomicAdd(o + 0, a * v.x);
    unsafeAtomicAdd(o + 1, a * v.y);
    unsafeAtomicAdd(o + 2, a * v.z);
    unsafeAtomicAdd(o + 3, a * v.w);
}

// ---------- P6: normalize + bias ----------
__global__ void k_finalize(float* __restrict__ out, const float* __restrict__ sden,
                           const float* __restrict__ bias, int Nn) {
    long long idx = (long long)blockIdx.x * blockDim.x + threadIdx.x;
    long long tot = (long long)Nn * OUT_C;
    if (idx >= tot) return;
    int i = (int)(idx >> 7);
    int c = (int)(idx & (OUT_C - 1));
    out[idx] = out[idx] / (sden[i] + 1e-16f) + bias[c];
}

extern "C" void kernel_launch(void* const* d_in, const int* in_sizes, int n_in,
                              void* d_out, int out_size, void* d_ws, size_t ws_size,
                              hipStream_t stream) {
    const float* x    = (const float*)d_in[0];
    const int*   ei   = (const int*)  d_in[1];
    const float* W    = (const float*)d_in[2];
    const float* att  = (const float*)d_in[3];
    const float* bias = (const float*)d_in[4];
    float* out = (float*)d_out;

    const int Nn   = in_sizes[0] / IN_C;   // 50000
    const int Etot = in_sizes[1] / 2;      // 850000
    const int* src = ei;                   // edge_index[0]
    const int* dst = ei + Etot;            // edge_index[1]

    // workspace layout (all 16B-aligned given hipMalloc base alignment)
    char* ws = (char*)d_ws;
    float*    h    = (float*)ws;                          ws += (size_t)Nn * OUT_C * sizeof(float);
    float*    al   = (float*)ws;                          ws += (size_t)Nn * sizeof(float);
    float*    ar   = (float*)ws;                          ws += (size_t)Nn * sizeof(float);
    float*    sden = (float*)ws;                          ws += (size_t)Nn * sizeof(float);
    unsigned* menc = (unsigned*)ws;                       ws += (size_t)Nn * sizeof(unsigned);
    float*    aE   = (float*)ws;                          ws += (size_t)Etot * sizeof(float);

    const int B = 256;
    long long outElems = (long long)Nn * OUT_C;

    // P0: init
    k_init<<<(unsigned)((outElems + B - 1) / B), B, 0, stream>>>(out, sden, menc, Nn);

    // P1: WMMA GEMM (8 waves/block, one 16x64 strip per wave)
    int mTiles = (Nn + 15) / 16;
    long long waves = (long long)mTiles * 2;
    k_gemm_wmma<<<(unsigned)((waves + 7) / 8), B, 0, stream>>>(x, W, h, Nn);

    // P2: node scores (1 wave/node)
    k_scores<<<(unsigned)(((long long)Nn + 7) / 8), B, 0, stream>>>(h, att, al, ar, Nn);

    // P3: edge logits + segment max
    k_edge_alpha<<<(Etot + B - 1) / B, B, 0, stream>>>(al, ar, src, dst, aE, menc, Etot);

    // P4: exp + segment sum
    k_edge_exp<<<(Etot + B - 1) / B, B, 0, stream>>>(dst, aE, menc, sden, Etot);

    // P5: scatter aggregation (1 wave/edge, 8 waves/block)
    k_edge_scatter<<<(unsigned)(((long long)Etot + 7) / 8), B, 0, stream>>>(
        h, aE, src, dst, out, Etot);

    // P6: normalize + bias
    k_finalize<<<(unsigned)((outElems + B - 1) / B), B, 0, stream>>>(out, sden, bias, Nn);
}